// GAT_90211493085598
// MI455X (gfx1250) — compile-verified
//
#include <hip/hip_runtime.h>
#include <cmath>

// ---------------------------------------------------------------------------
// GAT (2x GATConv + MLP + segment pool + p1^T p2) for MI455X / gfx1250.
// GEMMs via V_WMMA_F32_16X16X4_F32 (full f32, wave32), B staged in LDS,
// NCOL templated so the inner loop is pure imm-offset loads + WMMA.
// Edge softmax + aggregation via L2-resident f32 atomics (node tables fit
// in the 192MB L2). Pooling hierarchically through LDS (batch is sorted).
// ---------------------------------------------------------------------------

typedef float v2f __attribute__((ext_vector_type(2)));
typedef float v8f __attribute__((ext_vector_type(8)));

#define HDIM 128
#define CDIM 64
#define GDIM 64
#define NEG_SLOPE 0.2f

// ---------------- GEMM: C[M,NCOL] = A[M,128] @ B[128,NCOL] (+epilogue) ------
// mode 0: raw store; mode 1: +bias, ReLU; mode 2: +bias.
// grid.x = M/16 (M multiple of 16). blockDim.x = 32*(NCOL/16).
template <int NCOL>
__global__ void gat_gemm_wmma_f32(const float* __restrict__ A,
                                  const float* __restrict__ B,
                                  float* __restrict__ Cm,
                                  const float* __restrict__ bias,
                                  int mode) {
  __shared__ float Bs[HDIM * NCOL];
  // cooperative stage of the whole B panel into LDS (float4 = b128 loads)
  const int nb4 = (HDIM * NCOL) / 4;
  for (int i = threadIdx.x; i < nb4; i += blockDim.x)
    ((float4*)Bs)[i] = ((const float4*)B)[i];
  __syncthreads();

  const int lane = threadIdx.x & 31;
  const int wave = threadIdx.x >> 5;
  const int row  = blockIdx.x * 16 + (lane & 15);
  const int koff = (lane >> 4) << 1;          // lanes 0-15 -> K{0,1}; 16-31 -> K{2,3}
  const int col  = wave * 16 + (lane & 15);

  v8f acc = {};
  const float* __restrict__ arow = A + (size_t)row * HDIM + koff;
  const float* __restrict__ brow = Bs + (size_t)koff * NCOL + col;
#pragma unroll
  for (int k = 0; k < HDIM; k += 4) {
    v2f a, b;
    const float2 av = *(const float2*)(arow + k);   // imm-offset global_load_b64
    a.x = av.x;
    a.y = av.y;
    b.x = brow[k * NCOL];                           // imm-offset ds_load_b32
    b.y = brow[k * NCOL + NCOL];
    acc = __builtin_amdgcn_wmma_f32_16x16x4_f32(false, a, false, b,
                                                (short)0, acc, false, false);
  }

  const float bv = (mode > 0) ? bias[col] : 0.0f;
#pragma unroll
  for (int r = 0; r < 8; ++r) {
    const int mr = blockIdx.x * 16 + r + ((lane >> 4) << 3);  // C/D VGPR layout
    float v = acc[r] + bv;
    if (mode == 1) v = v > 0.0f ? v : 0.0f;
    Cm[(size_t)mr * NCOL + col] = v;
  }
}

// ---------------- per-node attention dots: alpha = h . att ------------------
__global__ void gat_row_dots(const float* __restrict__ h,
                             const float* __restrict__ a_s,
                             const float* __restrict__ a_d,
                             float* __restrict__ out_s,
                             float* __restrict__ out_d, int n) {
  const int lane = threadIdx.x & 31;
  const int row = (blockIdx.x * blockDim.x + threadIdx.x) >> 5;
  if (row >= n) return;
  const float* hr = h + (size_t)row * HDIM;
  float ss = 0.f, sd = 0.f;
#pragma unroll
  for (int j = 0; j < 4; ++j) {
    const int idx = lane + 32 * j;
    const float v = hr[idx];
    ss += v * a_s[idx];
    sd += v * a_d[idx];
  }
#pragma unroll
  for (int off = 16; off; off >>= 1) {
    ss += __shfl_xor(ss, off, 32);
    sd += __shfl_xor(sd, off, 32);
  }
  if (lane == 0) { out_s[row] = ss; out_d[row] = sd; }
}

// ---------------- fills -----------------------------------------------------
__global__ void gat_fill4(float4* __restrict__ p, float v, int n4) {
  const int i = blockIdx.x * blockDim.x + threadIdx.x;
  if (i < n4) p[i] = make_float4(v, v, v, v);
}

// ---------------- edge kernels ----------------------------------------------
__device__ inline float atomic_max_f32(float* addr, float val) {
  if (val >= 0.0f)
    return __int_as_float(atomicMax((int*)addr, __float_as_int(val)));
  return __uint_as_float(atomicMin((unsigned int*)addr, __float_as_uint(val)));
}

__device__ inline void edge_endpoints(const int* __restrict__ ei, int t, int E_,
                                      int& src, int& dst) {
  if (t < E_) { src = ei[t]; dst = ei[E_ + t]; }
  else        { src = t - E_; dst = t - E_; }      // self-loops appended
}

__global__ void gat_edge_max(const int* __restrict__ ei,
                             const float* __restrict__ as,
                             const float* __restrict__ ad,
                             float* __restrict__ mx, int E_, int n) {
  const int t = blockIdx.x * blockDim.x + threadIdx.x;
  if (t >= E_ + n) return;
  int src, dst; edge_endpoints(ei, t, E_, src, dst);
  float s = as[src] + ad[dst];
  s = (s > 0.f) ? s : NEG_SLOPE * s;
  atomic_max_f32(&mx[dst], s);
}

__global__ void gat_edge_expsum(const int* __restrict__ ei,
                                const float* __restrict__ as,
                                const float* __restrict__ ad,
                                const float* __restrict__ mx,
                                float* __restrict__ denom, int E_, int n) {
  const int t = blockIdx.x * blockDim.x + threadIdx.x;
  if (t >= E_ + n) return;
  int src, dst; edge_endpoints(ei, t, E_, src, dst);
  float s = as[src] + ad[dst];
  s = (s > 0.f) ? s : NEG_SLOPE * s;
  atomicAdd(&denom[dst], __expf(s - mx[dst]));
}

// one wave per edge; each lane moves float4 (32 lanes * 4 = 128 features)
__global__ void gat_aggregate(const int* __restrict__ ei,
                              const float* __restrict__ as,
                              const float* __restrict__ ad,
                              const float* __restrict__ mx,
                              const float* __restrict__ denom,
                              const float* __restrict__ h,
                              float* __restrict__ outp, int E_, int n) {
  const int lane = threadIdx.x & 31;
  const int edge = (blockIdx.x * blockDim.x + threadIdx.x) >> 5;
  if (edge >= E_ + n) return;
  int src, dst; edge_endpoints(ei, edge, E_, src, dst);
  float s = as[src] + ad[dst];
  s = (s > 0.f) ? s : NEG_SLOPE * s;
  const float a = __expf(s - mx[dst]) / denom[dst];
  const float4 hv = ((const float4*)(h + (size_t)src * HDIM))[lane];
  float* op = outp + (size_t)dst * HDIM + lane * 4;
  atomicAdd(op + 0, a * hv.x);
  atomicAdd(op + 1, a * hv.y);
  atomicAdd(op + 2, a * hv.z);
  atomicAdd(op + 3, a * hv.w);
}

__global__ void gat_bias_elu(float* __restrict__ x, const float* __restrict__ b,
                             int n) {
  const int i = blockIdx.x * blockDim.x + threadIdx.x;
  if (i >= n) return;
  const float v = x[i] + b[i & (HDIM - 1)];
  x[i] = v > 0.f ? v : (__expf(v) - 1.0f);
}

// ---------------- pooling: p[batch[i]] += y[i] ; batch sorted ---------------
#define POOL_NODES 256
__global__ void gat_pool(const float* __restrict__ y,
                         const int* __restrict__ batch,
                         float* __restrict__ p, int n) {
  __shared__ float acc[GDIM * CDIM];           // 16 KB of the 320 KB LDS
  for (int i = threadIdx.x; i < GDIM * CDIM; i += blockDim.x) acc[i] = 0.f;
  __syncthreads();
  const int base = blockIdx.x * POOL_NODES;
  const int end = min(base + POOL_NODES, n);
  const int c = threadIdx.x & 63;
  const int sub = threadIdx.x >> 6;            // 4 node-lanes x 64 cols
  for (int i = base + sub; i < end; i += 4) {
    const int g = batch[i];
    atomicAdd(&acc[g * CDIM + c], y[(size_t)i * CDIM + c]);
  }
  __syncthreads();
  // batch is sorted: this block only touched graphs [gmin, gmax]
  const int gmin = batch[base];
  const int gmax = batch[end - 1];
  const int cnt = (gmax - gmin + 1) * CDIM;
  for (int idx = threadIdx.x; idx < cnt; idx += blockDim.x)
    atomicAdd(&p[gmin * CDIM + idx], acc[gmin * CDIM + idx]);
}

// ---------------- final: out[r,c] = sum_g p1[g,r]*p2[g,c] -------------------
__global__ void gat_final(const float* __restrict__ p1,
                          const float* __restrict__ p2,
                          float* __restrict__ out) {
  const int t = blockIdx.x * blockDim.x + threadIdx.x;  // 4096
  const int r = t >> 6, c = t & 63;
  float s = 0.f;
#pragma unroll 8
  for (int g = 0; g < GDIM; ++g) s += p1[g * CDIM + r] * p2[g * CDIM + c];
  out[t] = s;
}

// ---------------------------------------------------------------------------
extern "C" void kernel_launch(void* const* d_in, const int* in_sizes, int n_in,
                              void* d_out, int out_size, void* d_ws, size_t ws_size,
                              hipStream_t stream) {
  const float* x1      = (const float*)d_in[0];
  const float* x2      = (const float*)d_in[1];
  const float* convW   = (const float*)d_in[2];   // [2,128,128]
  const float* att_src = (const float*)d_in[3];   // [2,128]
  const float* att_dst = (const float*)d_in[4];   // [2,128]
  const float* conv_b  = (const float*)d_in[5];   // [2,128]
  const float* mlp_w1  = (const float*)d_in[6];   // [128,128]
  const float* mlp_b1  = (const float*)d_in[7];   // [128]
  const float* mlp_w2  = (const float*)d_in[8];   // [128,64]
  const float* mlp_b2  = (const float*)d_in[9];   // [64]
  const int*   ei1     = (const int*)d_in[10];    // [2,E]
  const int*   batch1  = (const int*)d_in[11];    // [N] sorted
  const int*   ei2     = (const int*)d_in[12];
  const int*   batch2  = (const int*)d_in[13];

  const int N = in_sizes[0] / HDIM;               // 50000
  const int E = in_sizes[10] / 2;                 // 500000
  const int Etot = E + N;

  // ---- workspace layout (~52 MB) ----
  char* ws = (char*)d_ws;
  const size_t szNH = (size_t)N * HDIM * sizeof(float);
  float* bufA = (float*)(ws);                     // h / z1
  float* bufB = (float*)(ws + szNH);              // aggregated out / y
  float* as_  = (float*)(ws + 2 * szNH);
  float* ad_  = as_ + N;
  float* mx   = ad_ + N;
  float* dn   = mx + N;
  float* p1   = dn + N;
  float* p2   = p1 + GDIM * CDIM;

  const int BLK = 256;
  const int gemmGrid  = N / 16;                       // 3125, exact
  const int dotGrid   = (N * 32 + BLK - 1) / BLK;
  const int edgeGrid  = (Etot + BLK - 1) / BLK;
  const int aggGrid   = (Etot * 32 + BLK - 1) / BLK;  // wave per edge
  const int nhGrid    = (N * HDIM + BLK - 1) / BLK;
  const int nh4Grid   = (N * HDIM / 4 + BLK - 1) / BLK;
  const int n4Grid    = (N / 4 + BLK - 1) / BLK;
  const int poolGrid  = (N + POOL_NODES - 1) / POOL_NODES;

  for (int br = 0; br < 2; ++br) {
    const float* x     = br ? x2 : x1;
    const int*   ei    = br ? ei2 : ei1;
    const int*   batch = br ? batch2 : batch1;
    float*       p     = br ? p2 : p1;

    const float* in = x;
    for (int l = 0; l < 2; ++l) {
      // h = in @ W_l  (f32 WMMA, B in LDS)
      gat_gemm_wmma_f32<HDIM><<<gemmGrid, 256, 0, stream>>>(
          in, convW + (size_t)l * HDIM * HDIM, bufA, nullptr, 0);
      // alpha_s / alpha_d
      gat_row_dots<<<dotGrid, BLK, 0, stream>>>(
          bufA, att_src + l * HDIM, att_dst + l * HDIM, as_, ad_, N);
      // reset max / denom / output accumulator
      gat_fill4<<<n4Grid, BLK, 0, stream>>>((float4*)mx, -INFINITY, N / 4);
      gat_fill4<<<n4Grid, BLK, 0, stream>>>((float4*)dn, 0.0f, N / 4);
      gat_fill4<<<nh4Grid, BLK, 0, stream>>>((float4*)bufB, 0.0f, N * HDIM / 4);
      // segment softmax + weighted aggregation
      gat_edge_max<<<edgeGrid, BLK, 0, stream>>>(ei, as_, ad_, mx, E, N);
      gat_edge_expsum<<<edgeGrid, BLK, 0, stream>>>(ei, as_, ad_, mx, dn, E, N);
      gat_aggregate<<<aggGrid, BLK, 0, stream>>>(ei, as_, ad_, mx, dn,
                                                 bufA, bufB, E, N);
      gat_bias_elu<<<nhGrid, BLK, 0, stream>>>(bufB, conv_b + l * HDIM, N * HDIM);
      in = bufB;
    }

    // MLP: z1 = relu(bufB @ w1 + b1) ; y = z1 @ w2 + b2
    gat_gemm_wmma_f32<HDIM><<<gemmGrid, 256, 0, stream>>>(bufB, mlp_w1, bufA,
                                                          mlp_b1, 1);
    gat_gemm_wmma_f32<CDIM><<<gemmGrid, 128, 0, stream>>>(bufA, mlp_w2, bufB,
                                                          mlp_b2, 2);
    // segment-sum pool
    gat_fill4<<<(GDIM * CDIM / 4 + BLK - 1) / BLK, BLK, 0, stream>>>(
        (float4*)p, 0.0f, GDIM * CDIM / 4);
    gat_pool<<<poolGrid, BLK, 0, stream>>>(bufB, batch, p, N);
  }

  gat_final<<<(GDIM * CDIM) / BLK, BLK, 0, stream>>>(p1, p2, (float*)d_out);
}